// SubtractionLayer_29772713296023
// MI455X (gfx1250) — compile-verified
//
#include <hip/hip_runtime.h>

// SubtractionLayer: out = whole_h; out[lig_idx] -= alpha*lig_h; out[rec_idx] -= beta*rec_h.
// lig_idx and rec_idx together cover every row of whole_h exactly once, so the
// whole thing is two disjoint streaming passes — pure HBM-bandwidth bound
// (~786 MB moved, ~0.17 FLOP/byte). Strategy: float4 (B128) fully-coalesced
// traffic with non-temporal hints, one float4 per lane, 4 waves per 512-float row.

typedef float f32x4 __attribute__((ext_vector_type(4)));

static constexpr int D  = 512;      // hidden dim (floats per row)
static constexpr int DV = D / 4;    // float4 vectors per row (128)

__global__ __launch_bounds__(256) void scatter_sub_rows_kernel(
    const f32x4* __restrict__ whole,   // N x DV   (row-major)
    const f32x4* __restrict__ part,    // nrows x DV
    const int*   __restrict__ idx,     // nrows    (unique row ids into whole/out)
    const float* __restrict__ scale,   // 1
    f32x4*       __restrict__ out,     // N x DV
    int nrows)
{
    long long t     = (long long)blockIdx.x * blockDim.x + threadIdx.x;
    long long total = (long long)nrows * DV;
    if (t >= total) return;

    int row = (int)(t >> 7);          // t / DV  (DV == 128)
    int col = (int)(t & (DV - 1));    // t % DV

    float s = scale[0];               // uniform scalar (alpha or beta)

    long long w = (long long)idx[row] * DV + col;   // destination vector index

    // Streamed exactly once -> non-temporal (TH=NT) loads/stores, B128 each.
    f32x4 a = __builtin_nontemporal_load(&whole[w]);
    f32x4 b = __builtin_nontemporal_load(&part[t]);

    f32x4 r;
    r.x = a.x - s * b.x;
    r.y = a.y - s * b.y;
    r.z = a.z - s * b.z;
    r.w = a.w - s * b.w;

    __builtin_nontemporal_store(r, &out[w]);
}

extern "C" void kernel_launch(void* const* d_in, const int* in_sizes, int n_in,
                              void* d_out, int out_size, void* d_ws, size_t ws_size,
                              hipStream_t stream) {
    // setup_inputs() order:
    //   0: whole_h (N*D f32), 1: lig_h (B*L*D f32), 2: rec_h (B*R*D f32),
    //   3: alpha (1 f32), 4: beta (1 f32), 5: lig_idx (B*L i32), 6: rec_idx (B*R i32)
    const float* whole   = (const float*)d_in[0];
    const float* lig     = (const float*)d_in[1];
    const float* rec     = (const float*)d_in[2];
    const float* alpha   = (const float*)d_in[3];
    const float* beta    = (const float*)d_in[4];
    const int*   lig_idx = (const int*)d_in[5];
    const int*   rec_idx = (const int*)d_in[6];
    float*       out     = (float*)d_out;

    const int lig_rows = in_sizes[1] / D;   // B*L = 51200
    const int rec_rows = in_sizes[2] / D;   // B*R = 76800

    const int block = 256;                  // 8 waves (wave32) per block
    const long long lig_vec = (long long)lig_rows * DV;
    const long long rec_vec = (long long)rec_rows * DV;
    const int lig_blocks = (int)((lig_vec + block - 1) / block);
    const int rec_blocks = (int)((rec_vec + block - 1) / block);

    // The two index sets partition the rows of out, so the two launches write
    // disjoint rows and every output row is written exactly once (no copy pass).
    scatter_sub_rows_kernel<<<lig_blocks, block, 0, stream>>>(
        (const f32x4*)whole, (const f32x4*)lig, lig_idx, alpha, (f32x4*)out, lig_rows);
    scatter_sub_rows_kernel<<<rec_blocks, block, 0, stream>>>(
        (const f32x4*)whole, (const f32x4*)rec, rec_idx, beta, (f32x4*)out, rec_rows);
}